// AdaConv_49830210568595
// MI455X (gfx1250) — compile-verified
//
#include <hip/hip_runtime.h>
#include <hip/hip_bf16.h>

typedef float v2f __attribute__((ext_vector_type(2)));
typedef float v4f __attribute__((ext_vector_type(4)));
typedef float v8f __attribute__((ext_vector_type(8)));

#define NB 4
#define NC 64
#define NH 128
#define NW 128
#define NE 16
#define NP (NH * NW)   // 16384
#define NJ 576         // 64*9
#define CSTR 18        // LDS channel-fastest stride (16 chans + 2 pad floats)

// ---------------- Kernel 1: para0 = relu(conv1x1(x, pw) + pb) ----------------
__global__ void __launch_bounds__(256) k_conv1x1_relu(
    const float* __restrict__ x, const float* __restrict__ pw,
    const float* __restrict__ pb, float* __restrict__ out0) {
  __shared__ float spw[NE * NC];
  __shared__ float spb[NE];
  const int tid = threadIdx.x;
  for (int i = tid; i < NE * NC; i += 256) spw[i] = pw[i];
  if (tid < NE) spb[tid] = pb[tid];
  __syncthreads();

  const int p = blockIdx.x * 256 + tid;
  const int b = p >> 14;
  const int pi = p & (NP - 1);
  const float* xb = x + (size_t)b * NC * NP + pi;

  float acc[NE];
#pragma unroll
  for (int e = 0; e < NE; ++e) acc[e] = spb[e];
  for (int c = 0; c < NC; ++c) {
    const float xv = xb[(size_t)c * NP];
#pragma unroll
    for (int e = 0; e < NE; ++e) acc[e] = fmaf(spw[e * NC + c], xv, acc[e]);
  }
  float* ob = out0 + (size_t)b * NE * NP + pi;
#pragma unroll
  for (int e = 0; e < NE; ++e) ob[(size_t)e * NP] = fmaxf(acc[e], 0.0f);
}

// ---------------- Kernel 2: para = conv3x3(para0, cw) + cb ----------------
__global__ void __launch_bounds__(256) k_conv3x3(
    const float* __restrict__ in0, const float* __restrict__ cw,
    const float* __restrict__ cb, float* __restrict__ para) {
  __shared__ float scw[NE * NE * 9];
  __shared__ float scb[NE];
  const int tid = threadIdx.x;
  for (int i = tid; i < NE * NE * 9; i += 256) scw[i] = cw[i];
  if (tid < NE) scb[tid] = cb[tid];
  __syncthreads();

  const int p = blockIdx.x * 256 + tid;
  const int b = p >> 14;
  const int pi = p & (NP - 1);
  const int h = pi >> 7, w = pi & (NW - 1);
  const float* ib = in0 + (size_t)b * NE * NP;

  float acc[NE];
#pragma unroll
  for (int e = 0; e < NE; ++e) acc[e] = scb[e];
  for (int ep = 0; ep < NE; ++ep) {
#pragma unroll
    for (int kh = 0; kh < 3; ++kh) {
      const int hs = h + kh - 1;
      if ((unsigned)hs >= (unsigned)NH) continue;
#pragma unroll
      for (int kw = 0; kw < 3; ++kw) {
        const int ws = w + kw - 1;
        if ((unsigned)ws >= (unsigned)NW) continue;
        const float v = ib[(size_t)ep * NP + hs * NW + ws];
#pragma unroll
        for (int e = 0; e < NE; ++e)
          acc[e] = fmaf(scw[(e * NE + ep) * 9 + kh * 3 + kw], v, acc[e]);
      }
    }
  }
  float* ob = para + (size_t)b * NE * NP + pi;
#pragma unroll
  for (int e = 0; e < NE; ++e) ob[(size_t)e * NP] = acc[e];
}

// ------------- Kernel 2b: pack W into per-lane WMMA A fragments -------------
// K permutation: step t = chunk*36 + khkw*4 + cg covers channels
// c0 = chunk*16 + cg*4 .. c0+3 at fixed khkw; j(kk) = (c0+kk)*9 + khkw.
// Fragment for lane (hi,ln): a.x = Wmat[o*16+ln, j(2hi)], a.y = Wmat[.., j(2hi+1)]
// with Wmat[o*16+e, j] = W[(o*576+j)*16 + e].
__global__ void __launch_bounds__(256) k_packA(
    const float* __restrict__ W, float* __restrict__ Apack) {
  const int gid = blockIdx.x * 256 + threadIdx.x;   // (o*144 + t)*32 + lane
  const int lane = gid & 31;
  const int t = (gid >> 5) % 144;
  const int o = (gid >> 5) / 144;
  const int hi = lane >> 4, ln = lane & 15;
  const int chunk = t / 36, s = t % 36;
  const int khkw = s >> 2, cg = s & 3;
  const int c0 = chunk * 16 + cg * 4;
  const int j0 = (c0 + 2 * hi) * 9 + khkw;
  v2f a;
  a.x = W[((size_t)(o * NJ + j0)) * NE + ln];
  a.y = W[((size_t)(o * NJ + j0 + 9)) * NE + ln];
  *(v2f*)(Apack + (size_t)gid * 2) = a;
}

// ---------------- Kernel 3: main WMMA GEMM + epsilon reduction ----------------
// grid = (B*H, 8); 8 waves; wave -> output channel o. Accumulates
// t[o*16+e, p] (e = M rows) for a full 128-pixel row (8 N-tiles).
__global__ void __launch_bounds__(256) k_adaconv_main(
    const float* __restrict__ x, const float* __restrict__ Apack,
    const float* __restrict__ para, float* __restrict__ out) {
  __shared__ float sx2[3 * 130 * CSTR];   // [kh][col 0..129][cc], 28080B
  __shared__ float spara[NE * NW];        // 8192B

  const int tid = threadIdx.x;
  const int lane = tid & 31;
  const int wv = tid >> 5;
  const int hi = lane >> 4;
  const int ln = lane & 15;
  const int row = blockIdx.x;             // b*128 + h
  const int b = row >> 7, h = row & (NH - 1);
  const int o = blockIdx.y * 8 + wv;

  // stage para row (16 x 128)
  {
    const float* pbase = para + (size_t)b * NE * NP + h * NW;
    for (int i = tid; i < 512; i += 256) {
      const int e = i >> 5, q = i & 31;
      v4f v = *(const v4f*)(pbase + (size_t)e * NP + q * 4);
      *(v4f*)(spara + e * NW + q * 4) = v;
    }
  }

  v8f acc[8];
#pragma unroll
  for (int nt = 0; nt < 8; ++nt)
#pragma unroll
    for (int r = 0; r < 8; ++r) acc[nt][r] = 0.0f;

  for (int chunk = 0; chunk < 4; ++chunk) {
    __syncthreads();
    // ---- stage 16-channel chunk, transposed channel-fastest ----
    const float* xb = x + (size_t)(b * NC + chunk * 16) * NP;
    for (int i = tid; i < 48 * 32; i += 256) {
      const int r = i >> 5, q = i & 31;   // r = cc*3 + kh
      const int cc = r / 3, kh = r - cc * 3;
      const int hs = h + kh - 1;
      v4f v = {0.0f, 0.0f, 0.0f, 0.0f};
      if ((unsigned)hs < (unsigned)NH)
        v = *(const v4f*)(xb + (size_t)cc * NP + hs * NW + q * 4);
      const int base = (kh * 130 + q * 4 + 1) * CSTR + cc;
      sx2[base] = v[0];
      sx2[base + CSTR] = v[1];
      sx2[base + 2 * CSTR] = v[2];
      sx2[base + 3 * CSTR] = v[3];
    }
    for (int i = tid; i < 96; i += 256) {  // zero W-halo columns 0 and 129
      const int kh = i >> 5, rem = i & 31;
      const int cc = rem & 15, edge = rem >> 4;
      sx2[(kh * 130 + (edge ? 129 : 0)) * CSTR + cc] = 0.0f;
    }
    __syncthreads();

    // ---- K loop: 9 (kh,kw) x 4 channel-groups, K=4 per WMMA ----
    const float* pA = Apack + ((size_t)(o * 144 + chunk * 36) * 32 + lane) * 2;
#pragma unroll 1
    for (int khkw = 0; khkw < 9; ++khkw) {
      const int kh = khkw / 3, kw = khkw - kh * 3;
      // b64 pair (cc, cc+1): value x[c0+2hi+{0,1}][h+kh-1][p+kw-1]
      const float* bbase = sx2 + (kh * 130 + ln + kw) * CSTR + 2 * hi;
#pragma unroll
      for (int cg = 0; cg < 4; ++cg) {
        const v2f a = *(const v2f*)(pA + (khkw * 4 + cg) * 64);
#pragma unroll
        for (int nt = 0; nt < 8; ++nt) {
          const v2f bb = *(const v2f*)(bbase + cg * 4 + nt * 16 * CSTR);
          acc[nt] = __builtin_amdgcn_wmma_f32_16x16x4_f32(
              false, a, false, bb, (short)0, acc[nt], false, false);
        }
      }
    }
  }

  // ---- epsilon reduction: out[o,p] = sum_e acc[e,p] * para[e,p] ----
  float* ob = out + ((size_t)(b * NC + o) * NH + h) * NW;
#pragma unroll
  for (int nt = 0; nt < 8; ++nt) {
    const int p = nt * 16 + ln;
    float partial = 0.0f;
#pragma unroll
    for (int r = 0; r < 8; ++r)
      partial = fmaf(acc[nt][r], spara[(r + 8 * hi) * NW + p], partial);
    const float tot = partial + __shfl_xor(partial, 16, 32);
    if (hi == 0) ob[p] = tot;
  }
}

extern "C" void kernel_launch(void* const* d_in, const int* in_sizes, int n_in,
                              void* d_out, int out_size, void* d_ws, size_t ws_size,
                              hipStream_t stream) {
  const float* x  = (const float*)d_in[0];   // (4,64,128,128)
  const float* W  = (const float*)d_in[1];   // (1,1,36864,16)
  const float* pw = (const float*)d_in[2];   // (16,64,1,1)
  const float* pb = (const float*)d_in[3];   // (16,)
  const float* cw = (const float*)d_in[4];   // (16,16,3,3)
  const float* cb = (const float*)d_in[5];   // (16,)
  float* out = (float*)d_out;                // (4,64,128,128)

  float* ws0   = (float*)d_ws;                    // relu(1x1), later reused for Apack
  float* para  = ws0 + (size_t)NB * NE * NP;      // +4MB: predictor output
  float* Apack = ws0;                             // 64*144*32*2 f32 = 2.36MB (reuse)

  const int pix_blocks = (NB * NP) / 256;         // 256
  k_conv1x1_relu<<<pix_blocks, 256, 0, stream>>>(x, pw, pb, ws0);
  k_conv3x3<<<pix_blocks, 256, 0, stream>>>(ws0, cw, cb, para);
  k_packA<<<(64 * 144 * 32) / 256, 256, 0, stream>>>(W, Apack);  // 1152 blocks

  dim3 grid(NB * NH, 8);                          // (512, 8)
  k_adaconv_main<<<grid, 256, 0, stream>>>(x, Apack, para, out);
}